// GNNDecoder_9715216023655
// MI455X (gfx1250) — compile-verified
//
#include <hip/hip_runtime.h>
#include <hip/hip_bf16.h>

// GNNDecoder for MI455X (gfx1250, wave32).
//
// Key optimization: graph_embedding has only B=4096 distinct rows, so the
// N=500k / E=1M GEMMs collapse to two tiny per-graph GEMMs (4096x256x128 and
// 4096x256x64, ~0.4 GFLOP, done with V_WMMA_F32_16X16X4_F32) followed by a
// pure gather/scatter of the 2MB/1MB result tables (L2-resident on a 192MB L2)
// into 512MB of streaming output, written with non-temporal B128 stores.
// Roofline: ~520MB total traffic / 23.3 TB/s ~= 22us, write-bandwidth bound.

typedef __attribute__((ext_vector_type(2))) float v2f;
typedef __attribute__((ext_vector_type(4))) float v4f;
typedef __attribute__((ext_vector_type(8))) float v8f;

// ---------------------------------------------------------------------------
// C[M x Nn] = A[M x K] @ W[Nn x K]^T + bias[Nn]   (both row-major, "NT" GEMM)
// One wave32 computes one 16x16 tile of C via V_WMMA_F32_16X16X4_F32.
// Requires M,Nn multiples of 16 and K multiple of 4 (holds: 4096/128/64/256).
// ---------------------------------------------------------------------------
__global__ __launch_bounds__(256)
void gemm_nt_wmma_f32(const float* __restrict__ A,
                      const float* __restrict__ W,
                      const float* __restrict__ bias,
                      float* __restrict__ C,
                      int M, int Nn, int K)
{
    const int lane   = threadIdx.x & 31;
    const int waveId = blockIdx.x * (blockDim.x >> 5) + (threadIdx.x >> 5);

    const int tilesN = Nn >> 4;
    const int tileM  = waveId / tilesN;
    const int tileN  = waveId - tileM * tilesN;
    // Wave-uniform exit: EXEC stays all-1s for waves that reach the WMMA.
    if (tileM >= (M >> 4)) return;

    const int m0   = tileM << 4;
    const int n0   = tileN << 4;
    const int half = lane >> 4;       // 0: lanes 0-15, 1: lanes 16-31
    const int l15  = lane & 15;
    const int koff = half << 1;       // A/B frags: lanes 0-15 hold K={0,1},
                                      //            lanes 16-31 hold K={2,3}
    const float* __restrict__ Arow = A + (size_t)(m0 + l15) * K + koff;
    const float* __restrict__ Wrow = W + (size_t)(n0 + l15) * K + koff;

    v8f c = {};
    #pragma unroll 4
    for (int k = 0; k < K; k += 4) {
        v2f a = *(const v2f*)(Arow + k);   // 8B-aligned (k,koff even)
        v2f b = *(const v2f*)(Wrow + k);
        // 8 args: (neg_a, A, neg_b, B, c_mod, C, reuse_a, reuse_b)
        c = __builtin_amdgcn_wmma_f32_16x16x4_f32(
                false, a, false, b, (short)0, c, false, false);
    }

    // C/D layout: VGPR r -> row m0 + r + 8*half, col n0 + l15.
    const float bv   = bias[n0 + l15];
    const int   mrow = m0 + (half << 3);
    float* __restrict__ Cp = C + (size_t)mrow * Nn + (n0 + l15);
    #pragma unroll
    for (int r = 0; r < 8; ++r)
        Cp[(size_t)r * Nn] = c[r] + bv;
}

// ---------------------------------------------------------------------------
// x_hat[n][:] = pgNode[batch[n]][:]   (FN = 128 floats = 32 lanes x float4)
// One wave per node; non-temporal stores keep the 256MB stream out of caches.
// ---------------------------------------------------------------------------
__global__ __launch_bounds__(256)
void scatter_nodes(const int* __restrict__ batch,
                   const float* __restrict__ pgNode,  // B x FN
                   float* __restrict__ outX,          // N x FN
                   int N, int FN)
{
    const int lane = threadIdx.x & 31;
    const int n    = blockIdx.x * (blockDim.x >> 5) + (threadIdx.x >> 5);
    if (n >= N) return;

    const int g = batch[n];                          // broadcast load, L2 hit
    const v4f* __restrict__ src = (const v4f*)(pgNode + (size_t)g * FN);
    v4f* __restrict__       dst = (v4f*)(outX + (size_t)n * FN);
    const int nv4 = FN >> 2;                         // 32 for FN=128
    for (int i = lane; i < nv4; i += 32)
        __builtin_nontemporal_store(src[i], dst + i);
}

// ---------------------------------------------------------------------------
// edge_hat[e][:] = pgEdge[batch[edge_src[e]]][:]  (FE = 64 floats)
// Two edges per wave: 16 lanes x float4 each.
// ---------------------------------------------------------------------------
__global__ __launch_bounds__(256)
void scatter_edges(const int* __restrict__ batch,
                   const int* __restrict__ edge_src,  // edge_index row 0
                   const float* __restrict__ pgEdge,  // B x FE
                   float* __restrict__ outE,          // E x FE
                   int E, int FE)
{
    const int lane   = threadIdx.x & 31;
    const int waveId = blockIdx.x * (blockDim.x >> 5) + (threadIdx.x >> 5);
    const int e      = waveId * 2 + (lane >> 4);
    if (e >= E) return;

    const int l15 = lane & 15;
    const int s   = edge_src[e];
    const int g   = batch[s];
    const v4f* __restrict__ src = (const v4f*)(pgEdge + (size_t)g * FE);
    v4f* __restrict__       dst = (v4f*)(outE + (size_t)e * FE);
    const int nv4 = FE >> 2;                         // 16 for FE=64
    for (int i = l15; i < nv4; i += 16)
        __builtin_nontemporal_store(src[i], dst + i);
}

extern "C" void kernel_launch(void* const* d_in, const int* in_sizes, int n_in,
                              void* d_out, int out_size, void* d_ws, size_t ws_size,
                              hipStream_t stream)
{
    // Inputs (setup_inputs order):
    //  0: graph_embedding [B,H] f32   1: batch [N] int   2: edge_index [2,E] int
    //  3: node_w [FN,H] f32  4: node_b [FN] f32  5: edge_w [FE,H] f32  6: edge_b [FE]
    const float* GE      = (const float*)d_in[0];
    const int*   batch   = (const int*)d_in[1];
    const int*   eidx    = (const int*)d_in[2];
    const float* node_w  = (const float*)d_in[3];
    const float* node_b  = (const float*)d_in[4];
    const float* edge_w  = (const float*)d_in[5];
    const float* edge_b  = (const float*)d_in[6];

    const int FN = in_sizes[4];               // 128
    const int FE = in_sizes[6];               // 64
    const int H  = in_sizes[3] / FN;          // 256
    const int B  = in_sizes[0] / H;           // 4096
    const int N  = in_sizes[1];               // 500000
    const int E  = in_sizes[2] / 2;           // 1000000

    float* outX = (float*)d_out;              // [N, FN]
    float* outE = outX + (size_t)N * FN;      // [E, FE]

    float* pgNode = (float*)d_ws;             // [B, FN]  (2 MB)
    float* pgEdge = pgNode + (size_t)B * FN;  // [B, FE]  (1 MB)

    // Stage 1: per-graph GEMMs via WMMA (one 16x16 tile per wave, 8 waves/block)
    {
        int tiles  = (B >> 4) * (FN >> 4);    // 2048
        int blocks = (tiles + 7) >> 3;
        gemm_nt_wmma_f32<<<blocks, 256, 0, stream>>>(GE, node_w, node_b,
                                                     pgNode, B, FN, H);
    }
    {
        int tiles  = (B >> 4) * (FE >> 4);    // 1024
        int blocks = (tiles + 7) >> 3;
        gemm_nt_wmma_f32<<<blocks, 256, 0, stream>>>(GE, edge_w, edge_b,
                                                     pgEdge, B, FE, H);
    }

    // Stage 2: bandwidth-bound scatter of the L2-resident tables.
    {
        int blocks = (N + 7) >> 3;            // 1 wave per node, 8 waves/block
        scatter_nodes<<<blocks, 256, 0, stream>>>(batch, pgNode, outX, N, FN);
    }
    {
        int waves  = (E + 1) >> 1;            // 2 edges per wave
        int blocks = (waves + 7) >> 3;
        scatter_edges<<<blocks, 256, 0, stream>>>(batch, eidx, pgEdge, outE, E, FE);
    }
}